// get_model_15814069584234
// MI455X (gfx1250) — compile-verified
//
#include <hip/hip_runtime.h>
#include <hip/hip_bf16.h>

typedef __attribute__((ext_vector_type(2))) float v2f;
typedef __attribute__((ext_vector_type(8))) float v8f;

#define BATCH 32
#define NPTS  4096

__device__ __forceinline__ constexpr float inv_sqrt_bn() {
    return 0.9999950000374996f; // 1/sqrt(1 + 1e-5)
}

// ---------------------------------------------------------------------------
// WMMA MLP layer: act_out[rows][COUT] = relu((act_in[rows][KPAD] * W^T + b)*g/sqrt(1+eps) + bt)
// rows = 16 * (blockDim.x/32); each wave owns one 16-row m-tile.
// If FINAL: instead of storing, fold ReLU-max over rows into LDS int pmax[COUT]
// (float bits, valid compare since post-ReLU >= 0).
// ---------------------------------------------------------------------------
template<int KPAD, int CIN, int COUT, bool FINAL>
__device__ __forceinline__ void mlp_layer(const float* __restrict__ act_in,
                                          float* __restrict__ act_out,
                                          int* __restrict__ pmax,
                                          const float* __restrict__ W,
                                          const float* __restrict__ bias,
                                          const float* __restrict__ gam,
                                          const float* __restrict__ bet)
{
    const int lane = threadIdx.x & 31;
    const int wave = threadIdx.x >> 5;
    const int half = lane >> 4;   // 0: K pair {0,1}; 1: K pair {2,3} within k-step
    const int lp   = lane & 15;
    const int arow = wave * 16 + lp;

    for (int nt = 0; nt < COUT / 16; ++nt) {
        const int n = nt * 16 + lp;
        v8f acc = {0.f, 0.f, 0.f, 0.f, 0.f, 0.f, 0.f, 0.f};
#pragma unroll
        for (int ks = 0; ks < KPAD / 4; ++ks) {
            const int k = ks * 4 + 2 * half;
            v2f a, b;
            a.x = act_in[arow * KPAD + k];
            a.y = act_in[arow * KPAD + k + 1];
            b.x = (k     < CIN) ? W[n * CIN + k]     : 0.0f;
            b.y = (k + 1 < CIN) ? W[n * CIN + k + 1] : 0.0f;
            acc = __builtin_amdgcn_wmma_f32_16x16x4_f32(
                false, a, false, b, (short)0, acc, false, false);
        }
        const float bb = bias[n];
        const float sc = gam[n] * inv_sqrt_bn();
        const float sh = bet[n];
        if (!FINAL) {
#pragma unroll
            for (int v = 0; v < 8; ++v) {
                const int m = wave * 16 + half * 8 + v;
                float val = (acc[v] + bb) * sc + sh;
                act_out[m * COUT + n] = fmaxf(val, 0.0f);
            }
        } else {
            float vm = 0.0f; // ReLU floor
#pragma unroll
            for (int v = 0; v < 8; ++v) {
                float val = (acc[v] + bb) * sc + sh;
                vm = fmaxf(vm, val);
            }
            atomicMax(&pmax[n], __float_as_int(vm));
        }
    }
}

// ---------------------------------------------------------------------------
// (B,3,N) -> (B,N,3)
// ---------------------------------------------------------------------------
__global__ void transpose_kernel(const float* __restrict__ in, float* __restrict__ out, int total)
{
    int t = blockIdx.x * blockDim.x + threadIdx.x;
    if (t >= total) return;
    int b = t / (NPTS * 3);
    int r = t - b * (NPTS * 3);
    int n = r / 3;
    int c = r - n * 3;
    out[t] = in[(b * 3 + c) * NPTS + n];
}

// ---------------------------------------------------------------------------
// Farthest point sampling: one block per batch, matches lax.scan semantics
// (records 'far' at iteration entry; far0 = 0). Also gathers new_xyz.
// ---------------------------------------------------------------------------
template<int N, int NPOINT>
__global__ void __launch_bounds__(256) fps_kernel(const float* __restrict__ pts,
                                                  int* __restrict__ fidx,
                                                  float* __restrict__ newxyz)
{
    constexpr int BLK = 256;
    constexpr int PPT = N / BLK;
    const int b = blockIdx.x;
    const int tid = threadIdx.x;
    __shared__ float dist[N];
    __shared__ float redv[BLK];
    __shared__ int   redi[BLK];
    __shared__ int   fhist[NPOINT];

    const float* P = pts + (size_t)b * N * 3;
    for (int i = tid; i < N; i += BLK) dist[i] = 1e10f;
    int far = 0;
    __syncthreads();

    for (int it = 0; it < NPOINT; ++it) {
        if (tid == 0) fhist[it] = far;
        const float cx = P[far * 3 + 0];
        const float cy = P[far * 3 + 1];
        const float cz = P[far * 3 + 2];
        float bv = -1.0f; int bi = 0;
        const int i0 = tid * PPT;
#pragma unroll
        for (int j = 0; j < PPT; ++j) {
            const int i = i0 + j;
            const float dx = P[i * 3 + 0] - cx;
            const float dy = P[i * 3 + 1] - cy;
            const float dz = P[i * 3 + 2] - cz;
            const float d = dx * dx + dy * dy + dz * dz;
            float dd = fminf(dist[i], d);
            dist[i] = dd;
            if (dd > bv) { bv = dd; bi = i; } // strict > keeps lowest index (first occurrence)
        }
        redv[tid] = bv; redi[tid] = bi;
        __syncthreads();
        for (int s = BLK / 2; s > 0; s >>= 1) {
            if (tid < s) {
                float ov = redv[tid + s]; int oi = redi[tid + s];
                float mv = redv[tid];     int mi = redi[tid];
                if (ov > mv || (ov == mv && oi < mi)) { redv[tid] = ov; redi[tid] = oi; }
            }
            __syncthreads();
        }
        far = redi[0];
        __syncthreads();
    }

    for (int j = tid; j < NPOINT; j += BLK) {
        int id = fhist[j];
        fidx[b * NPOINT + j] = id;
        newxyz[((size_t)b * NPOINT + j) * 3 + 0] = P[id * 3 + 0];
        newxyz[((size_t)b * NPOINT + j) * 3 + 1] = P[id * 3 + 1];
        newxyz[((size_t)b * NPOINT + j) * 3 + 2] = P[id * 3 + 2];
    }
}

// ---------------------------------------------------------------------------
// Ball query: first K points (index order) with |p - c|^2 <= r^2, pad with first.
// Distance computed as |s|^2 + |d|^2 - 2 s.d to mirror square_distance().
// ---------------------------------------------------------------------------
__global__ void ballquery_kernel(const float* __restrict__ pts,      // (B,N,3)
                                 const float* __restrict__ centers,  // (B,S,3)
                                 int* __restrict__ gidx,             // (B,S,K)
                                 int N, int S, int K, float r2, int total)
{
    int t = blockIdx.x * blockDim.x + threadIdx.x;
    if (t >= total) return;
    int b = t / S;
    const float* P = pts + (size_t)b * N * 3;
    const float cx = centers[(size_t)t * 3 + 0];
    const float cy = centers[(size_t)t * 3 + 1];
    const float cz = centers[(size_t)t * 3 + 2];
    const float cs = cx * cx + cy * cy + cz * cz;
    int* out = gidx + (size_t)t * K;
    int cnt = 0; int first = 0;
    for (int i = 0; i < N && cnt < K; ++i) {
        const float x = P[i * 3 + 0], y = P[i * 3 + 1], z = P[i * 3 + 2];
        const float sqr = cs + (x * x + y * y + z * z) - 2.0f * (cx * x + cy * y + cz * z);
        if (!(sqr > r2)) {
            if (cnt == 0) first = i;
            out[cnt++] = i;
        }
    }
    for (; cnt < K; ++cnt) out[cnt] = first;
}

// ---------------------------------------------------------------------------
// SA1 MLP: one block = one (b,s) group of 32 samples, 3ch -> 64 -> 64 -> 128, max over 32.
// ---------------------------------------------------------------------------
__global__ void __launch_bounds__(64) mlp1_kernel(
    const float* __restrict__ pts, const float* __restrict__ newxyz,
    const int* __restrict__ gidx, float* __restrict__ l1pts,
    const float* W0, const float* b0, const float* g0, const float* t0,
    const float* W1, const float* b1, const float* g1, const float* t1,
    const float* W2, const float* b2, const float* g2, const float* t2)
{
    __shared__ float act0[32 * 4];
    __shared__ float a1[32 * 64];
    __shared__ float a2[32 * 64];
    __shared__ int   pmax[128];
    const int t = threadIdx.x;
    const int g = blockIdx.x;           // b*512 + s
    const int b = g >> 9;
    const float cx = newxyz[(size_t)g * 3 + 0];
    const float cy = newxyz[(size_t)g * 3 + 1];
    const float cz = newxyz[(size_t)g * 3 + 2];
    if (t < 32) {
        int pi = gidx[(size_t)g * 32 + t];
        const float* P = pts + ((size_t)b * NPTS + pi) * 3;
        act0[t * 4 + 0] = P[0] - cx;
        act0[t * 4 + 1] = P[1] - cy;
        act0[t * 4 + 2] = P[2] - cz;
        act0[t * 4 + 3] = 0.0f;
    }
    for (int c = t; c < 128; c += 64) pmax[c] = 0;
    __syncthreads();
    mlp_layer<4, 3, 64, false>(act0, a1, nullptr, W0, b0, g0, t0);
    __syncthreads();
    mlp_layer<64, 64, 64, false>(a1, a2, nullptr, W1, b1, g1, t1);
    __syncthreads();
    mlp_layer<64, 64, 128, true>(a2, nullptr, pmax, W2, b2, g2, t2);
    __syncthreads();
    for (int c = t; c < 128; c += 64)
        l1pts[(size_t)g * 128 + c] = __int_as_float(pmax[c]);
}

// ---------------------------------------------------------------------------
// SA2 MLP: one block = half (32 of 64 samples) of one (b,s) group.
// 131ch (xyz diff + 128 feats) -> 128 -> 128 -> 256, max combined via global atomicMax.
// ---------------------------------------------------------------------------
__global__ void __launch_bounds__(64) mlp2_kernel(
    const float* __restrict__ l1xyz,   // (B,512,3)
    const float* __restrict__ l2xyz,   // (B,128,3)
    const int* __restrict__ gidx,      // (B,128,64)
    const float* __restrict__ l1pts,   // (B,512,128)
    int* __restrict__ l2pts,           // (B,128,256) accumulated as float bits, pre-zeroed
    const float* W0, const float* b0, const float* g0, const float* t0,
    const float* W1, const float* b1, const float* g1, const float* t1,
    const float* W2, const float* b2, const float* g2, const float* t2)
{
    __shared__ float in2[32 * 132];
    __shared__ float c1[32 * 128];
    __shared__ float c2[32 * 128];
    __shared__ int   pmax[256];
    const int t = threadIdx.x;
    const int h  = blockIdx.x & 1;       // which half of the 64 samples
    const int cs = blockIdx.x >> 1;      // b*128 + s
    const int b  = cs >> 7;
    for (int e = t; e < 32 * 132; e += 64) {
        const int r = e / 132;
        const int c = e - r * 132;
        const int k = h * 32 + r;
        const int pi = gidx[(size_t)cs * 64 + k];
        float val;
        if (c < 3)        val = l1xyz[((size_t)b * 512 + pi) * 3 + c] - l2xyz[(size_t)cs * 3 + c];
        else if (c < 131) val = l1pts[((size_t)b * 512 + pi) * 128 + (c - 3)];
        else              val = 0.0f;
        in2[e] = val;
    }
    for (int c = t; c < 256; c += 64) pmax[c] = 0;
    __syncthreads();
    mlp_layer<132, 131, 128, false>(in2, c1, nullptr, W0, b0, g0, t0);
    __syncthreads();
    mlp_layer<128, 128, 128, false>(c1, c2, nullptr, W1, b1, g1, t1);
    __syncthreads();
    mlp_layer<128, 128, 256, true>(c2, nullptr, pmax, W2, b2, g2, t2);
    __syncthreads();
    for (int c = t; c < 256; c += 64)
        atomicMax(&l2pts[(size_t)cs * 256 + c], pmax[c]);
}

// ---------------------------------------------------------------------------
// SA3 (group-all) MLP: one block (1 wave) = one 16-point tile of one batch.
// 259ch (xyz + 256 feats) -> 256 -> 512 -> 1024, max over 128 points via global atomicMax.
// ---------------------------------------------------------------------------
__global__ void __launch_bounds__(32) mlp3_kernel(
    const float* __restrict__ l2xyz,   // (B,128,3)
    const float* __restrict__ l2pts,   // (B,128,256)
    int* __restrict__ outp,            // (B,1024) as float bits, pre-zeroed
    const float* W0, const float* b0, const float* g0, const float* t0,
    const float* W1, const float* b1, const float* g1, const float* t1,
    const float* W2, const float* b2, const float* g2, const float* t2)
{
    __shared__ float big0[16 * 512];   // holds input (stride 260) then layer2 out (stride 512)
    __shared__ float mid[16 * 256];
    __shared__ int   pmax[1024];
    const int t = threadIdx.x;
    const int b  = blockIdx.x >> 3;
    const int mt = blockIdx.x & 7;
    for (int e = t; e < 16 * 260; e += 32) {
        const int r = e / 260;
        const int c = e - r * 260;
        const int p = mt * 16 + r;
        float val;
        if (c < 3)        val = l2xyz[((size_t)b * 128 + p) * 3 + c];
        else if (c < 259) val = l2pts[((size_t)b * 128 + p) * 256 + (c - 3)];
        else              val = 0.0f;
        big0[r * 260 + c] = val;
    }
    for (int c = t; c < 1024; c += 32) pmax[c] = 0;
    __syncthreads();
    mlp_layer<260, 259, 256, false>(big0, mid, nullptr, W0, b0, g0, t0);
    __syncthreads();
    mlp_layer<256, 256, 512, false>(mid, big0, nullptr, W1, b1, g1, t1);
    __syncthreads();
    mlp_layer<512, 512, 1024, true>(big0, nullptr, pmax, W2, b2, g2, t2);
    __syncthreads();
    for (int c = t; c < 1024; c += 32)
        atomicMax(&outp[(size_t)b * 1024 + c], pmax[c]);
}

// ---------------------------------------------------------------------------
extern "C" void kernel_launch(void* const* d_in, const int* in_sizes, int n_in,
                              void* d_out, int out_size, void* d_ws, size_t ws_size,
                              hipStream_t stream)
{
    const float* xyz = (const float*)d_in[0];
    const float* W[3][3]; const float* Bb[3][3]; const float* Gg[3][3]; const float* Bt[3][3];
    for (int s = 0; s < 3; ++s)
        for (int i = 0; i < 3; ++i) {
            int base = 1 + s * 12 + i * 4;
            W[s][i]  = (const float*)d_in[base + 0];
            Bb[s][i] = (const float*)d_in[base + 1];
            Gg[s][i] = (const float*)d_in[base + 2];
            Bt[s][i] = (const float*)d_in[base + 3];
        }

    char* w = (char*)d_ws;
    float* pts   = (float*)(w + 0);          // 32*4096*3 f  (1,572,864 B)
    float* nx1   = (float*)(w + 1572864);    // 32*512*3  f  (196,608 B)
    int*   fidx1 = (int*)  (w + 1769472);    // 32*512    i  (65,536 B)
    int*   gidx1 = (int*)  (w + 1835008);    // 32*512*32 i  (2,097,152 B)
    float* l1pts = (float*)(w + 3932160);    // 32*512*128 f (8,388,608 B)
    int*   fidx2 = (int*)  (w + 12320768);   // 32*128    i  (16,384 B)
    float* nx2   = (float*)(w + 12337152);   // 32*128*3  f  (49,152 B)
    int*   gidx2 = (int*)  (w + 12386304);   // 32*128*64 i  (1,048,576 B)
    int*   l2pts = (int*)  (w + 13434880);   // 32*128*256 f (4,194,304 B) -> ends 17,629,184

    // zero the max-accumulators (0 bits == 0.0f; all accumulated values are post-ReLU >= 0)
    hipMemsetAsync(l2pts, 0, (size_t)32 * 128 * 256 * sizeof(int), stream);
    hipMemsetAsync(d_out, 0, (size_t)out_size * sizeof(float), stream);

    const int totT = BATCH * NPTS * 3;
    transpose_kernel<<<(totT + 255) / 256, 256, 0, stream>>>(xyz, pts, totT);

    fps_kernel<4096, 512><<<BATCH, 256, 0, stream>>>(pts, fidx1, nx1);

    ballquery_kernel<<<(BATCH * 512 + 255) / 256, 256, 0, stream>>>(
        pts, nx1, gidx1, 4096, 512, 32, 0.2f * 0.2f, BATCH * 512);

    mlp1_kernel<<<BATCH * 512, 64, 0, stream>>>(
        pts, nx1, gidx1, l1pts,
        W[0][0], Bb[0][0], Gg[0][0], Bt[0][0],
        W[0][1], Bb[0][1], Gg[0][1], Bt[0][1],
        W[0][2], Bb[0][2], Gg[0][2], Bt[0][2]);

    fps_kernel<512, 128><<<BATCH, 256, 0, stream>>>(nx1, fidx2, nx2);

    ballquery_kernel<<<(BATCH * 128 + 255) / 256, 256, 0, stream>>>(
        nx1, nx2, gidx2, 512, 128, 64, 0.4f * 0.4f, BATCH * 128);

    mlp2_kernel<<<BATCH * 128 * 2, 64, 0, stream>>>(
        nx1, nx2, gidx2, l1pts, l2pts,
        W[1][0], Bb[1][0], Gg[1][0], Bt[1][0],
        W[1][1], Bb[1][1], Gg[1][1], Bt[1][1],
        W[1][2], Bb[1][2], Gg[1][2], Bt[1][2]);

    mlp3_kernel<<<BATCH * 8, 32, 0, stream>>>(
        nx2, (const float*)l2pts, (int*)d_out,
        W[2][0], Bb[2][0], Gg[2][0], Bt[2][0],
        W[2][1], Bb[2][1], Gg[2][1], Bt[2][1],
        W[2][2], Bb[2][2], Gg[2][2], Bt[2][2]);
}